// MKMMDLoss_70248485093595
// MI455X (gfx1250) — compile-verified
//
#include <hip/hip_runtime.h>

// MK-MMD loss, B=256, D=256, n=512, KERNEL_NUM=5, KERNEL_MUL=2, FIX_SIGMA=None.
// Output: single f32 scalar.

#define B_SZ 256
#define D_SZ 256
#define N_TOT 512
#define N_PAIRS 1024          // 4 groups x 256 index pairs
#define WS_PARTIAL_OFF 16     // float offset of per-wave partials in workspace

typedef __attribute__((ext_vector_type(2))) float v2f;
typedef __attribute__((ext_vector_type(8))) float v8f;

// ---------------------------------------------------------------------------
// Kernel A: bandwidth via closed form.
//   S = sum_d (D-d) * [ 2n * sum_i x_{i,d}^2 - 2 * (sum_i x_{i,d})^2 ]
//   bw = S / (n^2 - n) / KERNEL_MUL^(KERNEL_NUM/2)   (= /4)
// Stores ws[0] = -1/bw.
// ---------------------------------------------------------------------------
__global__ void mmd_bandwidth_kernel(const float* __restrict__ src,
                                     const float* __restrict__ tgt,
                                     float* __restrict__ ws) {
  __shared__ float red[256];
  const int d = threadIdx.x;           // feature column, 0..255
  float sv = 0.f, sq = 0.f;
  for (int i = 0; i < B_SZ; ++i) {     // coalesced: thread d reads column d
    float x = src[i * D_SZ + d];
    sv += x; sq += x * x;
  }
  for (int i = 0; i < B_SZ; ++i) {
    float x = tgt[i * D_SZ + d];
    sv += x; sq += x * x;
  }
  float sd = 2.f * (float)N_TOT * sq - 2.f * sv * sv;   // sum_{i,j} diff_d^2
  red[d] = (float)(D_SZ - d) * sd;                      // cumsum weight
  __syncthreads();
  for (int off = 128; off > 0; off >>= 1) {
    if (d < off) red[d] += red[d + off];
    __syncthreads();
  }
  if (d == 0) {
    float S  = red[0];
    float bw = S * (1.f / (float)(N_TOT * (N_TOT - 1)));
    bw *= 0.25f;                       // /= KERNEL_MUL^(KERNEL_NUM//2) = 4
    ws[0] = -1.f / bw;
  }
}

// ---------------------------------------------------------------------------
// Kernel B: one wave32 per (i,j) pair. Cumsum over the 256 features done as a
// lower-triangular 16x16 matmul per 16-feature segment using
// V_WMMA_F32_16X16X4_F32 (4 K-chunks), then cross-segment prefix via lane
// shuffles of accumulator row 15. Each lane then evaluates the 5-kernel
// Gaussian sum for its 8 owned (segment, position) cells.
//
// Layouts (wave32 f32 WMMA):
//   A 16x4 : lane l -> M = l&15 ; vgpr r in {0,1}, half h=l>>4 -> K = r + 2h
//   B 4x16 : lane l -> N = l&15 ; vgpr r in {0,1}, half h=l>>4 -> K = r + 2h
//   C/D    : vgpr r, lane l -> M = r + 8*(l>>4), N = l&15
// ---------------------------------------------------------------------------
__global__ void mmd_pair_kernel(const float* __restrict__ src,
                                const float* __restrict__ tgt,
                                float* __restrict__ ws) {
  const int lane = threadIdx.x & 31;
  const int wave = threadIdx.x >> 5;
  const int p    = blockIdx.x * (blockDim.x >> 5) + wave;   // 0..1023

  const int g = p >> 8;          // pair group
  const int i = p & 255;
  const int j = (i + 1) & 255;

  int ra, rb; float sign;
  if (g == 0)      { ra = i;         rb = j;         sign =  1.f; } // k[i, j]
  else if (g == 1) { ra = i + B_SZ;  rb = j + B_SZ;  sign =  1.f; } // k[i+B, j+B]
  else if (g == 2) { ra = i;         rb = j + B_SZ;  sign = -1.f; } // k[i, j+B]
  else             { ra = j;         rb = i + B_SZ;  sign = -1.f; } // k[j, i+B]

  const float* __restrict__ rowA =
      (ra < B_SZ) ? (src + (size_t)ra * D_SZ) : (tgt + (size_t)(ra - B_SZ) * D_SZ);
  const float* __restrict__ rowB =
      (rb < B_SZ) ? (src + (size_t)rb * D_SZ) : (tgt + (size_t)(rb - B_SZ) * D_SZ);

  const int seg = lane & 15;     // B-operand column (= segment index), also A's M
  const int h   = lane >> 4;     // lane half selects K offset

  v8f acc = {};                  // O[M][seg] accumulator, starts at 0
#pragma unroll
  for (int s = 0; s < 4; ++s) {
    const int k = 4 * s + 2 * h; // this lane's first K row of the chunk
    // diff^2 values for B operand: V[k][seg] = (xa[seg*16+k] - xb[...])^2
    v2f xa = *(const v2f*)(rowA + seg * 16 + k);
    v2f xb = *(const v2f*)(rowB + seg * 16 + k);
    v2f bm;
    bm.x = (xa.x - xb.x) * (xa.x - xb.x);
    bm.y = (xa.y - xb.y) * (xa.y - xb.y);
    // lower-triangular L for A operand: L[M][k] = (k <= M)
    v2f am;
    am.x = (k     <= seg) ? 1.f : 0.f;
    am.y = (k + 1 <= seg) ? 1.f : 0.f;
    acc = __builtin_amdgcn_wmma_f32_16x16x4_f32(
        /*neg_a=*/false, am, /*neg_b=*/false, bm,
        /*c_mod=*/(short)0, acc, /*reuse_a=*/false, /*reuse_b=*/false);
  }

  // Segment totals live at M=15: vgpr 7, lanes 16..31 (lane 16+q holds seg q).
  // Exclusive prefix over segments -> per-lane offset for this lane's segment.
  float a7  = acc[7];
  float off = 0.f;
#pragma unroll
  for (int q = 0; q < 15; ++q) {
    float t = __shfl(a7, 16 + q, 32);
    if (q < seg) off += t;
  }

  // 5-bandwidth Gaussian kernel sum over this lane's 8 cumulative distances.
  const float ninv = ws[0];      // -1/bw
  float partial = 0.f;
#pragma unroll
  for (int r = 0; r < 8; ++r) {
    float c = acc[r] + off;      // full cumsum at d = seg*16 + (r + 8h)
    float t = c * ninv;          // t = -c/bw  (<= 0)
    partial += __expf(t)                 // bw * 2^0
             + __expf(t * 0.5f)         // bw * 2^1
             + __expf(t * 0.25f)        // bw * 2^2
             + __expf(t * 0.125f)       // bw * 2^3
             + __expf(t * 0.0625f);     // bw * 2^4
  }

  // Wave reduction; deterministic per-pair partial written to workspace.
#pragma unroll
  for (int o = 16; o > 0; o >>= 1) partial += __shfl_down(partial, o, 32);
  if (lane == 0) ws[WS_PARTIAL_OFF + p] = sign * partial;
}

// ---------------------------------------------------------------------------
// Kernel C: fixed-order reduction of the 1024 per-pair partials; scale by
// 1/(B*D) (loss/B then mean over D).
// ---------------------------------------------------------------------------
__global__ void mmd_finalize_kernel(const float* __restrict__ ws,
                                    float* __restrict__ out) {
  __shared__ float red[256];
  const int t = threadIdx.x;
  float s = 0.f;
#pragma unroll
  for (int q = 0; q < 4; ++q) s += ws[WS_PARTIAL_OFF + t * 4 + q];
  red[t] = s;
  __syncthreads();
  for (int off = 128; off > 0; off >>= 1) {
    if (t < off) red[t] += red[t + off];
    __syncthreads();
  }
  if (t == 0) out[0] = red[0] * (1.f / ((float)B_SZ * (float)D_SZ));
}

extern "C" void kernel_launch(void* const* d_in, const int* in_sizes, int n_in,
                              void* d_out, int out_size, void* d_ws, size_t ws_size,
                              hipStream_t stream) {
  const float* src = (const float*)d_in[0];
  const float* tgt = (const float*)d_in[1];
  float* ws  = (float*)d_ws;
  float* out = (float*)d_out;

  // Phase 1: bandwidth (closed form; one block).
  mmd_bandwidth_kernel<<<1, 256, 0, stream>>>(src, tgt, ws);
  // Phase 2: 1024 pair waves = 128 blocks x 8 waves.
  mmd_pair_kernel<<<N_PAIRS / 8, 256, 0, stream>>>(src, tgt, ws);
  // Phase 3: deterministic final reduction.
  mmd_finalize_kernel<<<1, 256, 0, stream>>>(ws, out);
}